// Net_721554506019
// MI455X (gfx1250) — compile-verified
//
#include <hip/hip_runtime.h>
#include <math.h>

// PointNet++ (SSG) forward for B=32 clouds of N=2048 points, f32 end-to-end.
// MLP matmuls use CDNA5 V_WMMA_F32_16X16X4_F32 (exact f32 matrix core).
// BN/relu are monotone (gamma>0, var>0 for these inputs) so the masked
// neighbor-max is done on raw WMMA accumulators, epilogue applied per channel.

typedef float v2f __attribute__((ext_vector_type(2)));
typedef float v8f __attribute__((ext_vector_type(8)));

#define NEGBIG (-3.402823466e38f)

__device__ __forceinline__ v8f wmma4(v2f a, v2f b, v8f c) {
  // D = A(16x4,f32) * B(4x16,f32) + C(16x16,f32)
  return __builtin_amdgcn_wmma_f32_16x16x4_f32(false, a, false, b, (short)0, c,
                                               false, false);
}

// ---------------------------------------------------------------------------
// Greedy farthest-point sampling, one block per cloud, cloud resident in LDS.
// Matches reference: start at index 0, mind=min(mind,d2(last)), argmax (first
// occurrence on ties).
// ---------------------------------------------------------------------------
__global__ void __launch_bounds__(256) fps_kernel(const float* __restrict__ pos,
                                                  float* __restrict__ centers,
                                                  int n, int m) {
  __shared__ float sPos[2048 * 3];
  __shared__ float sMind[2048];
  __shared__ float sRedV[256];
  __shared__ int sRedI[256];
  __shared__ int sLast;
  const int t = threadIdx.x;
  const float* p = pos + (size_t)blockIdx.x * n * 3;
  float* c = centers + (size_t)blockIdx.x * m * 3;
  for (int i = t; i < n * 3; i += 256) sPos[i] = p[i];
  for (int i = t; i < n; i += 256) sMind[i] = 1e10f;
  if (t == 0) sLast = 0;
  __syncthreads();
  if (t == 0) { c[0] = sPos[0]; c[1] = sPos[1]; c[2] = sPos[2]; }
  for (int it = 1; it < m; ++it) {
    const int last = sLast;
    const float lx = sPos[last * 3 + 0];
    const float ly = sPos[last * 3 + 1];
    const float lz = sPos[last * 3 + 2];
    float bv = -1.0f;
    int bi = 0x7fffffff;
    for (int q = t; q < n; q += 256) {
      float dx = sPos[q * 3 + 0] - lx;
      float dy = sPos[q * 3 + 1] - ly;
      float dz = sPos[q * 3 + 2] - lz;
      float d = dx * dx + dy * dy + dz * dz;
      float mn = fminf(sMind[q], d);
      sMind[q] = mn;
      if (mn > bv) { bv = mn; bi = q; }  // ascending q + strict > => first max
    }
    sRedV[t] = bv;
    sRedI[t] = bi;
    __syncthreads();
    for (int s = 128; s > 0; s >>= 1) {
      if (t < s) {
        float ov = sRedV[t + s];
        int oi = sRedI[t + s];
        if (ov > sRedV[t] || (ov == sRedV[t] && oi < sRedI[t])) {
          sRedV[t] = ov;
          sRedI[t] = oi;
        }
      }
      __syncthreads();
    }
    if (t == 0) {
      int nxt = sRedI[0];
      sLast = nxt;
      c[it * 3 + 0] = sPos[nxt * 3 + 0];
      c[it * 3 + 1] = sPos[nxt * 3 + 1];
      c[it * 3 + 2] = sPos[nxt * 3 + 2];
    }
    __syncthreads();
  }
}

// ---------------------------------------------------------------------------
// SA1: per center (one wave32 each): exact top-32 nearest of 2048 points,
// radius mask r^2=0.04, rel-pos MLP [32x3]@[3x32] via 4 f32 WMMAs, masked
// row-max, then bias/relu/BN per channel.
// ---------------------------------------------------------------------------
__global__ void __launch_bounds__(256) sa1_kernel(
    const float* __restrict__ pos, const float* __restrict__ c1,
    const float* __restrict__ pW1, const float* __restrict__ pb1,
    const float* __restrict__ pg1, const float* __restrict__ pbe1,
    const float* __restrict__ pm1, const float* __restrict__ pv1,
    float* __restrict__ h1) {
  __shared__ float sPos[2048 * 3];
  const int cloud = blockIdx.x >> 7;           // 128 blocks per cloud
  const int cbase = (blockIdx.x & 127) * 8;    // 8 centers (waves) per block
  const float* p = pos + (size_t)cloud * 2048 * 3;
  for (int i = threadIdx.x; i < 2048 * 3; i += 256) sPos[i] = p[i];
  __syncthreads();

  const int lane = threadIdx.x & 31;
  const int wave = threadIdx.x >> 5;
  const int ci = cbase + wave;
  const size_t cOff = ((size_t)cloud * 1024 + ci) * 3;
  const float cx = c1[cOff + 0], cy = c1[cOff + 1], cz = c1[cOff + 2];

  // --- top-32 selection: 32 rounds of wave-min with lowest-index tiebreak ---
  unsigned long long excl = 0ull;  // each lane owns points j*32+lane, j<64
  int myN = 0;
  float myD = 0.0f;
  for (int r = 0; r < 32; ++r) {
    float bd = 3.0e38f;
    int bp = 0x7fffffff;
    for (int j = 0; j < 64; ++j) {
      if ((excl >> j) & 1ull) continue;
      int pt = j * 32 + lane;
      float dx = sPos[pt * 3 + 0] - cx;
      float dy = sPos[pt * 3 + 1] - cy;
      float dz = sPos[pt * 3 + 2] - cz;
      float d = dx * dx + dy * dy + dz * dz;
      if (d < bd || (d == bd && pt < bp)) { bd = d; bp = pt; }
    }
    for (int off = 16; off > 0; off >>= 1) {
      float od = __shfl_xor(bd, off, 32);
      int op = __shfl_xor(bp, off, 32);
      if (od < bd || (od == bd && op < bp)) { bd = od; bp = op; }
    }
    if ((bp & 31) == lane) excl |= (1ull << (bp >> 5));
    if (lane == r) { myN = bp; myD = bd; }
  }

  const float relx = sPos[myN * 3 + 0] - cx;
  const float rely = sPos[myN * 3 + 1] - cy;
  const float relz = sPos[myN * 3 + 2] - cz;
  const bool valid = (myD <= 0.04f);  // r=0.2
  const unsigned vmask = (unsigned)(__ballot(valid) & 0xffffffffull);

  const bool hi = (lane >= 16);
  const int nlo = lane & 15;

  // B tiles of W1 (3x32, K padded to 4): VGPR k holds rows k (lanes 0-15)
  // and k+2 (lanes 16-31).
  v2f b0, b1;
  b0.x = hi ? pW1[2 * 32 + nlo] : pW1[nlo];
  b0.y = hi ? 0.0f : pW1[32 + nlo];
  b1.x = hi ? pW1[2 * 32 + 16 + nlo] : pW1[16 + nlo];
  b1.y = hi ? 0.0f : pW1[32 + 16 + nlo];

  // A tiles: lane=M row, VGPR+lanehalf=K. Row data gathered via shuffles.
  float x0 = __shfl(relx, nlo, 32), y0 = __shfl(rely, nlo, 32),
        z0 = __shfl(relz, nlo, 32);
  float x1 = __shfl(relx, nlo + 16, 32), y1 = __shfl(rely, nlo + 16, 32),
        z1 = __shfl(relz, nlo + 16, 32);
  v2f a0, a1;
  a0.x = hi ? z0 : x0;
  a0.y = hi ? 0.0f : y0;
  a1.x = hi ? z1 : x1;
  a1.y = hi ? 0.0f : y1;

  v8f c00 = {0, 0, 0, 0, 0, 0, 0, 0};
  v8f c01 = {0, 0, 0, 0, 0, 0, 0, 0};
  v8f c10 = {0, 0, 0, 0, 0, 0, 0, 0};
  v8f c11 = {0, 0, 0, 0, 0, 0, 0, 0};
  c00 = wmma4(a0, b0, c00);
  c01 = wmma4(a0, b1, c01);
  c10 = wmma4(a1, b0, c10);
  c11 = wmma4(a1, b1, c11);

  // Masked max over the 32 neighbor rows (C layout: VGPR e = rows e / e+8).
  float mx0 = NEGBIG, mx1 = NEGBIG;
#pragma unroll
  for (int e = 0; e < 8; ++e) {
    int r0 = e + (hi ? 8 : 0);
    int r1 = 16 + r0;
    bool ok0 = (vmask >> r0) & 1u;
    bool ok1 = (vmask >> r1) & 1u;
    mx0 = fmaxf(mx0, fmaxf(ok0 ? c00[e] : NEGBIG, ok1 ? c10[e] : NEGBIG));
    mx1 = fmaxf(mx1, fmaxf(ok0 ? c01[e] : NEGBIG, ok1 ? c11[e] : NEGBIG));
  }
  mx0 = fmaxf(mx0, __shfl_xor(mx0, 16, 32));
  mx1 = fmaxf(mx1, __shfl_xor(mx1, 16, 32));

  if (lane < 16) {
    float* o = h1 + ((size_t)cloud * 1024 + ci) * 32;
    float mv[2] = {mx0, mx1};
#pragma unroll
    for (int tg = 0; tg < 2; ++tg) {
      int ch = tg * 16 + lane;
      float v = mv[tg] + pb1[ch];
      v = fmaxf(v, 0.0f);
      v = (v - pm1[ch]) * rsqrtf(pv1[ch] + 1e-5f) * pg1[ch] + pbe1[ch];
      o[ch] = v;
    }
  }
}

// ---------------------------------------------------------------------------
// SA2: centers c2 (256/cloud) over c1 (1024 pts) with 32-ch features h1,
// r^2=0.16, MLP [32x35]@[35x64] via 9 k-steps x 8 WMMAs, masked max, epilogue.
// ---------------------------------------------------------------------------
__device__ __forceinline__ float feat35(const float* __restrict__ h1r, float rx,
                                        float ry, float rz, int k) {
  if (k < 32) return h1r[k];
  if (k == 32) return rx;
  if (k == 33) return ry;
  if (k == 34) return rz;
  return 0.0f;
}

__global__ void __launch_bounds__(256) sa2_kernel(
    const float* __restrict__ c1, const float* __restrict__ h1,
    const float* __restrict__ c2, const float* __restrict__ pW2,
    const float* __restrict__ pb2, const float* __restrict__ pg2,
    const float* __restrict__ pbe2, const float* __restrict__ pm2,
    const float* __restrict__ pv2, float* __restrict__ h2) {
  __shared__ float sPos[1024 * 3];
  const int cloud = blockIdx.x >> 5;         // 32 blocks per cloud
  const int cbase = (blockIdx.x & 31) * 8;   // 8 centers per block
  const float* p = c1 + (size_t)cloud * 1024 * 3;
  for (int i = threadIdx.x; i < 1024 * 3; i += 256) sPos[i] = p[i];
  __syncthreads();

  const int lane = threadIdx.x & 31;
  const int wave = threadIdx.x >> 5;
  const int ci = cbase + wave;
  const size_t cOff = ((size_t)cloud * 256 + ci) * 3;
  const float cx = c2[cOff + 0], cy = c2[cOff + 1], cz = c2[cOff + 2];

  unsigned excl = 0u;  // each lane owns points j*32+lane, j<32
  int myN = 0;
  float myD = 0.0f;
  for (int r = 0; r < 32; ++r) {
    float bd = 3.0e38f;
    int bp = 0x7fffffff;
    for (int j = 0; j < 32; ++j) {
      if ((excl >> j) & 1u) continue;
      int pt = j * 32 + lane;
      float dx = sPos[pt * 3 + 0] - cx;
      float dy = sPos[pt * 3 + 1] - cy;
      float dz = sPos[pt * 3 + 2] - cz;
      float d = dx * dx + dy * dy + dz * dz;
      if (d < bd || (d == bd && pt < bp)) { bd = d; bp = pt; }
    }
    for (int off = 16; off > 0; off >>= 1) {
      float od = __shfl_xor(bd, off, 32);
      int op = __shfl_xor(bp, off, 32);
      if (od < bd || (od == bd && op < bp)) { bd = od; bp = op; }
    }
    if ((bp & 31) == lane) excl |= (1u << (bp >> 5));
    if (lane == r) { myN = bp; myD = bd; }
  }

  const float relx = sPos[myN * 3 + 0] - cx;
  const float rely = sPos[myN * 3 + 1] - cy;
  const float relz = sPos[myN * 3 + 2] - cz;
  const bool valid = (myD <= 0.16f);  // r=0.4
  const unsigned vmask = (unsigned)(__ballot(valid) & 0xffffffffull);

  const bool hi = (lane >= 16);
  const int nlo = lane & 15;
  const int nb0 = __shfl(myN, nlo, 32);
  const int nb1 = __shfl(myN, nlo + 16, 32);
  const float rx0 = __shfl(relx, nlo, 32), ry0 = __shfl(rely, nlo, 32),
              rz0 = __shfl(relz, nlo, 32);
  const float rx1 = __shfl(relx, nlo + 16, 32), ry1 = __shfl(rely, nlo + 16, 32),
              rz1 = __shfl(relz, nlo + 16, 32);
  const float* h1r0 = h1 + ((size_t)cloud * 1024 + nb0) * 32;
  const float* h1r1 = h1 + ((size_t)cloud * 1024 + nb1) * 32;

  v8f acc0[4], acc1[4];
#pragma unroll
  for (int nt = 0; nt < 4; ++nt) {
    acc0[nt] = (v8f){0, 0, 0, 0, 0, 0, 0, 0};
    acc1[nt] = (v8f){0, 0, 0, 0, 0, 0, 0, 0};
  }
  for (int kk = 0; kk < 9; ++kk) {  // K=35 padded to 36
    int kx = kk * 4 + (hi ? 2 : 0);
    int ky = kx + 1;
    v2f a0, a1;
    a0.x = feat35(h1r0, rx0, ry0, rz0, kx);
    a0.y = feat35(h1r0, rx0, ry0, rz0, ky);
    a1.x = feat35(h1r1, rx1, ry1, rz1, kx);
    a1.y = feat35(h1r1, rx1, ry1, rz1, ky);
#pragma unroll
    for (int nt = 0; nt < 4; ++nt) {
      int col = nt * 16 + nlo;
      v2f b;
      b.x = (kx < 35) ? pW2[kx * 64 + col] : 0.0f;
      b.y = (ky < 35) ? pW2[ky * 64 + col] : 0.0f;
      acc0[nt] = wmma4(a0, b, acc0[nt]);
      acc1[nt] = wmma4(a1, b, acc1[nt]);
    }
  }

  float* o = h2 + ((size_t)cloud * 256 + ci) * 64;
#pragma unroll
  for (int nt = 0; nt < 4; ++nt) {
    float mx = NEGBIG;
#pragma unroll
    for (int e = 0; e < 8; ++e) {
      int r0 = e + (hi ? 8 : 0);
      int r1 = 16 + r0;
      mx = fmaxf(mx, fmaxf(((vmask >> r0) & 1u) ? acc0[nt][e] : NEGBIG,
                           ((vmask >> r1) & 1u) ? acc1[nt][e] : NEGBIG));
    }
    mx = fmaxf(mx, __shfl_xor(mx, 16, 32));
    if (lane < 16) {
      int ch = nt * 16 + lane;
      float v = mx + pb2[ch];
      v = fmaxf(v, 0.0f);
      v = (v - pm2[ch]) * rsqrtf(pv2[ch] + 1e-5f) * pg2[ch] + pbe2[ch];
      o[ch] = v;
    }
  }
}

// ---------------------------------------------------------------------------
// MLP3 + global max + classifier + log_softmax. One block (8 waves) per cloud:
// X[256,67] @ W3[67,128] via WMMA (each wave owns 32 rows), per-wave row-max
// into LDS, cross-wave max, bias/relu/BN, 128->10 classifier, log-softmax.
// ---------------------------------------------------------------------------
__device__ __forceinline__ float feat67(const float* __restrict__ h2r,
                                        const float* __restrict__ c2r, int k) {
  if (k < 64) return h2r[k];
  if (k < 67) return c2r[k - 64];
  return 0.0f;
}

__global__ void __launch_bounds__(256) mlp3_cls_kernel(
    const float* __restrict__ h2, const float* __restrict__ c2,
    const float* __restrict__ pW3, const float* __restrict__ pb3,
    const float* __restrict__ pg3, const float* __restrict__ pbe3,
    const float* __restrict__ pm3, const float* __restrict__ pv3,
    const float* __restrict__ pWc, const float* __restrict__ pbc,
    float* __restrict__ out) {
  __shared__ float sPart[8 * 128];
  __shared__ float sG[128];
  __shared__ float sL[10];
  const int cloud = blockIdx.x;
  const int lane = threadIdx.x & 31;
  const int wave = threadIdx.x >> 5;
  const bool hi = (lane >= 16);
  const int nlo = lane & 15;
  const float* h2b = h2 + (size_t)cloud * 256 * 64;
  const float* c2b = c2 + (size_t)cloud * 256 * 3;
  const int r0 = wave * 32 + nlo;       // M-tile 0 row for this lane
  const int r1 = r0 + 16;               // M-tile 1 row
  const float* h2r0 = h2b + r0 * 64;
  const float* h2r1 = h2b + r1 * 64;
  const float* c2r0 = c2b + r0 * 3;
  const float* c2r1 = c2b + r1 * 3;

  for (int nt = 0; nt < 8; ++nt) {
    v8f acc0 = {0, 0, 0, 0, 0, 0, 0, 0};
    v8f acc1 = {0, 0, 0, 0, 0, 0, 0, 0};
    for (int kk = 0; kk < 17; ++kk) {  // K=67 padded to 68
      int kx = kk * 4 + (hi ? 2 : 0);
      int ky = kx + 1;
      v2f a0, a1, b;
      a0.x = feat67(h2r0, c2r0, kx);
      a0.y = feat67(h2r0, c2r0, ky);
      a1.x = feat67(h2r1, c2r1, kx);
      a1.y = feat67(h2r1, c2r1, ky);
      int col = nt * 16 + nlo;
      b.x = (kx < 67) ? pW3[kx * 128 + col] : 0.0f;
      b.y = (ky < 67) ? pW3[ky * 128 + col] : 0.0f;
      acc0 = wmma4(a0, b, acc0);
      acc1 = wmma4(a1, b, acc1);
    }
    float mx = NEGBIG;
#pragma unroll
    for (int e = 0; e < 8; ++e) mx = fmaxf(mx, fmaxf(acc0[e], acc1[e]));
    mx = fmaxf(mx, __shfl_xor(mx, 16, 32));
    if (lane < 16) sPart[wave * 128 + nt * 16 + lane] = mx;
  }
  __syncthreads();
  if (threadIdx.x < 128) {
    int ch = threadIdx.x;
    float v = NEGBIG;
    for (int w = 0; w < 8; ++w) v = fmaxf(v, sPart[w * 128 + ch]);
    v += pb3[ch];
    v = fmaxf(v, 0.0f);
    v = (v - pm3[ch]) * rsqrtf(pv3[ch] + 1e-5f) * pg3[ch] + pbe3[ch];
    sG[ch] = v;
  }
  __syncthreads();
  if (threadIdx.x < 10) {
    float s = pbc[threadIdx.x];
    for (int k = 0; k < 128; ++k) s += sG[k] * pWc[k * 10 + threadIdx.x];
    sL[threadIdx.x] = fmaxf(s, 0.0f);
  }
  __syncthreads();
  if (threadIdx.x == 0) {
    float mx = sL[0];
    for (int i = 1; i < 10; ++i) mx = fmaxf(mx, sL[i]);
    float se = 0.0f;
    for (int i = 0; i < 10; ++i) se += expf(sL[i] - mx);
    float lse = mx + logf(se);
    for (int i = 0; i < 10; ++i) out[cloud * 10 + i] = sL[i] - lse;
  }
}

// ---------------------------------------------------------------------------
extern "C" void kernel_launch(void* const* d_in, const int* in_sizes, int n_in,
                              void* d_out, int out_size, void* d_ws,
                              size_t ws_size, hipStream_t stream) {
  (void)in_sizes; (void)n_in; (void)out_size; (void)ws_size;
  const float* pos = (const float*)d_in[0];
  const float* W1 = (const float*)d_in[1];
  const float* b1 = (const float*)d_in[2];
  const float* g1 = (const float*)d_in[3];
  const float* be1 = (const float*)d_in[4];
  const float* m1 = (const float*)d_in[5];
  const float* v1 = (const float*)d_in[6];
  const float* W2 = (const float*)d_in[7];
  const float* b2 = (const float*)d_in[8];
  const float* g2 = (const float*)d_in[9];
  const float* be2 = (const float*)d_in[10];
  const float* m2 = (const float*)d_in[11];
  const float* v2 = (const float*)d_in[12];
  const float* W3 = (const float*)d_in[13];
  const float* b3 = (const float*)d_in[14];
  const float* g3 = (const float*)d_in[15];
  const float* be3 = (const float*)d_in[16];
  const float* m3 = (const float*)d_in[17];
  const float* v3 = (const float*)d_in[18];
  const float* Wc = (const float*)d_in[19];
  const float* bc = (const float*)d_in[20];
  float* out = (float*)d_out;

  float* ws = (float*)d_ws;
  float* c1 = ws;                          // [32,1024,3]
  float* h1 = c1 + 32 * 1024 * 3;          // [32,1024,32]
  float* c2 = h1 + 32 * 1024 * 32;         // [32,256,3]
  float* h2 = c2 + 32 * 256 * 3;           // [32,256,64]

  fps_kernel<<<32, 256, 0, stream>>>(pos, c1, 2048, 1024);
  sa1_kernel<<<32 * 128, 256, 0, stream>>>(pos, c1, W1, b1, g1, be1, m1, v1, h1);
  fps_kernel<<<32, 256, 0, stream>>>(c1, c2, 1024, 256);
  sa2_kernel<<<32 * 32, 256, 0, stream>>>(c1, h1, c2, W2, b2, g2, be2, m2, v2, h2);
  mlp3_cls_kernel<<<32, 256, 0, stream>>>(h2, c2, W3, b3, g3, be3, m3, v3, Wc, bc, out);
}